// TGCN_Model_5282809774877
// MI455X (gfx1250) — compile-verified
//
#include <hip/hip_runtime.h>
#include <hip/hip_bf16.h>

typedef __attribute__((ext_vector_type(2))) float v2f;
typedef __attribute__((ext_vector_type(8))) float v8f;

#define HD 128            // H == D == 128
#define LDSS 132          // padded LDS row stride (conflict-free A-fragment reads)

__device__ __forceinline__ void atomicAddF(float* p, float v) {
  __hip_atomic_fetch_add(p, v, __ATOMIC_RELAXED, __HIP_MEMORY_SCOPE_AGENT);
}
__device__ __forceinline__ float sigmoidf(float x) {
  return 1.0f / (1.0f + __expf(-x));
}

// ---------------- init: deg = 1.0 (self-loop weight), pooled = 0 ----------------
__global__ void k_init(float* __restrict__ deg, float* __restrict__ pooled, int N) {
  int g = blockIdx.x * blockDim.x + threadIdx.x;
  if (g < N) deg[g] = 1.0f;
  if (g < HD) pooled[g] = 0.0f;
}

// ---------------- deg += w over edges ----------------
__global__ void k_deg(const int* __restrict__ dst, const float* __restrict__ w,
                      float* __restrict__ deg, int E) {
  int e = blockIdx.x * blockDim.x + threadIdx.x;
  if (e < E) atomicAddF(&deg[dst[e]], w[e]);
}

// ---------------- dinv = deg^-0.5 (in place) ----------------
__global__ void k_dinv(float* __restrict__ deg, int N) {
  int i = blockIdx.x * blockDim.x + threadIdx.x;
  if (i < N) {
    float d = deg[i];
    deg[i] = (d > 0.0f) ? (1.0f / sqrtf(d)) : 0.0f;
  }
}

// ---------------- agg init with self-loop term: agg[i][h] = dinv[i]^2 * X[i][h] ----------------
__global__ void k_selfloop(const float* __restrict__ X, const float* __restrict__ dinv,
                           float* __restrict__ agg, long total) {
  long g = (long)blockIdx.x * blockDim.x + threadIdx.x;
  if (g < total) {
    int i = (int)(g >> 7);
    float di = dinv[i];
    agg[g] = di * di * X[g];
  }
}

// ---------------- edge scatter: 32 lanes per edge, float4 per lane ----------------
__global__ void k_scatter(const float* __restrict__ X, const int* __restrict__ src,
                          const int* __restrict__ dst, const float* __restrict__ w,
                          const float* __restrict__ dinv, float* __restrict__ agg, int E) {
  long g = (long)blockIdx.x * blockDim.x + threadIdx.x;
  int e = (int)(g >> 5);
  if (e >= E) return;
  int lane = (int)(g & 31);
  int s = src[e], d = dst[e];
  float c = dinv[s] * w[e] * dinv[d];
  float4 x = ((const float4*)(X + (long)s * HD))[lane];
  float* ar = agg + (long)d * HD + lane * 4;
  atomicAddF(ar + 0, c * x.x);
  atomicAddF(ar + 1, c * x.y);
  atomicAddF(ar + 2, c * x.z);
  atomicAddF(ar + 3, c * x.w);
}

// ---------------- fold weights: M = W @ lW_top ; c = b @ lW_top + lb ----------------
// grid.x = 128 (row i), block = 128 (col j)
__global__ void k_prep(const float* __restrict__ W, const float* __restrict__ b,
                       const float* __restrict__ lW, const float* __restrict__ lb,
                       float* __restrict__ M, float* __restrict__ c) {
  int i = blockIdx.x, j = threadIdx.x;
  float acc = 0.0f;
  for (int k = 0; k < HD; ++k) acc += W[i * HD + k] * lW[k * HD + j];
  M[i * HD + j] = acc;
  if (i == 0) {
    float ca = 0.0f;
    for (int k = 0; k < HD; ++k) ca += b[k] * lW[k * HD + j];
    c[j] = ca + lb[j];
  }
}

// ---------------- WMMA fragment helpers (f32 16x16x4) ----------------
// A 16x4: lanes 0-15 hold row M=lane, K = k0..k0+1 ; lanes 16-31 hold K = k0+2..k0+3
// B 4x16: lanes 0-15 hold col N=lane, K = k0..k0+1 ; lanes 16-31 hold K = k0+2..k0+3
__device__ __forceinline__ v2f loadA(const float* sRow, int ks) {
  v2f a; a.x = sRow[ks]; a.y = sRow[ks + 1]; return a;
}
__device__ __forceinline__ v2f loadB(const float* __restrict__ W, int ks, int col) {
  v2f b; b.x = W[ks * HD + col]; b.y = W[(ks + 1) * HD + col]; return b;
}

// ---------------- gates: z = sigmoid(agg@Mz + h0@lzWb + cz), hr = h0 * sigmoid(...) ----------------
__global__ __launch_bounds__(256) void k_gates_zr(
    const float* __restrict__ agg, const float* __restrict__ h0,
    const float* __restrict__ Mz,  const float* __restrict__ Mr,
    const float* __restrict__ lzWb, const float* __restrict__ lrWb,
    const float* __restrict__ cz,  const float* __restrict__ cr,
    float* __restrict__ zOut, float* __restrict__ hrOut) {
  __shared__ float sA[16 * LDSS];
  __shared__ float sH[16 * LDSS];
  const int rowBase = blockIdx.x * 16;
  const int t = threadIdx.x;
  for (int idx = t; idx < 16 * HD; idx += 256) {
    int r = idx >> 7, c = idx & (HD - 1);
    sA[r * LDSS + c] = agg[(long)(rowBase + r) * HD + c];
    sH[r * LDSS + c] = h0[(long)(rowBase + r) * HD + c];
  }
  __syncthreads();
  const int wave = t >> 5;
  const int lane = t & 31;
  const int half = lane >> 4;
  const int rowA = lane & 15;
  const int colB = (wave << 4) + (lane & 15);
  v8f accZ = {}; v8f accR = {};
  for (int k0 = 0; k0 < HD; k0 += 4) {
    int ks = k0 + (half << 1);
    v2f aA = loadA(&sA[rowA * LDSS], ks);
    v2f aH = loadA(&sH[rowA * LDSS], ks);
    v2f bMz = loadB(Mz, ks, colB);
    v2f bLz = loadB(lzWb, ks, colB);
    v2f bMr = loadB(Mr, ks, colB);
    v2f bLr = loadB(lrWb, ks, colB);
    accZ = __builtin_amdgcn_wmma_f32_16x16x4_f32(false, aA, false, bMz, (short)0, accZ, false, false);
    accZ = __builtin_amdgcn_wmma_f32_16x16x4_f32(false, aH, false, bLz, (short)0, accZ, false, false);
    accR = __builtin_amdgcn_wmma_f32_16x16x4_f32(false, aA, false, bMr, (short)0, accR, false, false);
    accR = __builtin_amdgcn_wmma_f32_16x16x4_f32(false, aH, false, bLr, (short)0, accR, false, false);
  }
  const float czv = cz[colB], crv = cr[colB];
  for (int v = 0; v < 8; ++v) {
    int row = v + (half << 3);   // C/D layout: lanes 16-31 hold M+8
    float zval = sigmoidf(accZ[v] + czv);
    float rval = sigmoidf(accR[v] + crv);
    float h0v = sH[row * LDSS + colB];
    long o = (long)(rowBase + row) * HD + colB;
    zOut[o] = zval;
    hrOut[o] = h0v * rval;
  }
}

// ---------------- candidate + combine: h_t = tanh(agg@Mh + hr@lhWb + ch); h_new = z*h0+(1-z)*h_t ----------------
__global__ __launch_bounds__(256) void k_cand(
    const float* __restrict__ agg, const float* __restrict__ hr,
    const float* __restrict__ Mh, const float* __restrict__ lhWb,
    const float* __restrict__ ch, const float* __restrict__ zbuf,
    const float* __restrict__ h0, float* __restrict__ hnew) {
  __shared__ float sA[16 * LDSS];
  __shared__ float sR[16 * LDSS];
  const int rowBase = blockIdx.x * 16;
  const int t = threadIdx.x;
  for (int idx = t; idx < 16 * HD; idx += 256) {
    int r = idx >> 7, c = idx & (HD - 1);
    sA[r * LDSS + c] = agg[(long)(rowBase + r) * HD + c];
    sR[r * LDSS + c] = hr[(long)(rowBase + r) * HD + c];
  }
  __syncthreads();
  const int wave = t >> 5;
  const int lane = t & 31;
  const int half = lane >> 4;
  const int rowA = lane & 15;
  const int colB = (wave << 4) + (lane & 15);
  v8f accH = {};
  for (int k0 = 0; k0 < HD; k0 += 4) {
    int ks = k0 + (half << 1);
    v2f aA = loadA(&sA[rowA * LDSS], ks);
    v2f aR = loadA(&sR[rowA * LDSS], ks);
    v2f bMh = loadB(Mh, ks, colB);
    v2f bLh = loadB(lhWb, ks, colB);
    accH = __builtin_amdgcn_wmma_f32_16x16x4_f32(false, aA, false, bMh, (short)0, accH, false, false);
    accH = __builtin_amdgcn_wmma_f32_16x16x4_f32(false, aR, false, bLh, (short)0, accH, false, false);
  }
  const float chv = ch[colB];
  for (int v = 0; v < 8; ++v) {
    int row = v + (half << 3);
    long o = (long)(rowBase + row) * HD + colB;
    float ht = tanhf(accH[v] + chv);
    float z = zbuf[o];
    float h0v = h0[o];
    hnew[o] = z * h0v + (1.0f - z) * ht;
  }
}

// ---------------- pooled[h] += sum_u relu(h_new[ids[u]][h]) ----------------
__global__ void k_pool(const float* __restrict__ hnew, const int* __restrict__ ids,
                       float* __restrict__ pooled, int U) {
  int h = threadIdx.x;
  int u0 = blockIdx.x * 128;
  int u1 = min(u0 + 128, U);
  float acc = 0.0f;
  for (int u = u0; u < u1; ++u) {
    float v = hnew[(long)ids[u] * HD + h];
    acc += fmaxf(v, 0.0f);
  }
  atomicAddF(&pooled[h], acc);
}

// ---------------- head: z_graph = (pooled/U)@linW+linb; t=relu(zg@dnW+dnb); pred=sigmoid(t@outW+outb) ----------------
__global__ void k_head(const float* __restrict__ pooled,
                       const float* __restrict__ linW, const float* __restrict__ linb,
                       const float* __restrict__ dnW,  const float* __restrict__ dnb,
                       const float* __restrict__ outW, const float* __restrict__ outb,
                       float* __restrict__ pred, float invU, int C) {
  __shared__ float zg[HD];
  __shared__ float tb[HD];
  int j = threadIdx.x;
  float acc = 0.0f;
  for (int k = 0; k < HD; ++k) acc += (pooled[k] * invU) * linW[k * HD + j];
  zg[j] = acc + linb[j];
  __syncthreads();
  acc = 0.0f;
  for (int k = 0; k < HD; ++k) acc += zg[k] * dnW[k * HD + j];
  tb[j] = fmaxf(acc + dnb[j], 0.0f);
  __syncthreads();
  if (j < C) {
    acc = 0.0f;
    for (int k = 0; k < HD; ++k) acc += tb[k] * outW[k * C + j];
    pred[j] = sigmoidf(acc + outb[j]);
  }
}

extern "C" void kernel_launch(void* const* d_in, const int* in_sizes, int n_in,
                              void* d_out, int out_size, void* d_ws, size_t ws_size,
                              hipStream_t stream) {
  const float* node_feat  = (const float*)d_in[0];
  const float* edge_w     = (const float*)d_in[1];
  const float* h0         = (const float*)d_in[2];
  const float* Wz         = (const float*)d_in[3];
  const float* bz         = (const float*)d_in[4];
  const float* Wr         = (const float*)d_in[5];
  const float* br         = (const float*)d_in[6];
  const float* Wh         = (const float*)d_in[7];
  const float* bh         = (const float*)d_in[8];
  const float* lz_W       = (const float*)d_in[9];
  const float* lz_b       = (const float*)d_in[10];
  const float* lr_W       = (const float*)d_in[11];
  const float* lr_b       = (const float*)d_in[12];
  const float* lh_W       = (const float*)d_in[13];
  const float* lh_b       = (const float*)d_in[14];
  const float* lin_W      = (const float*)d_in[15];
  const float* lin_b      = (const float*)d_in[16];
  const float* dn_W       = (const float*)d_in[17];
  const float* dn_b       = (const float*)d_in[18];
  const float* out_W      = (const float*)d_in[19];
  const float* out_b      = (const float*)d_in[20];
  const int*   src        = (const int*)d_in[21];
  const int*   dst        = (const int*)d_in[22];
  const int*   local_nids = (const int*)d_in[23];

  const int N = in_sizes[0] / HD;
  const int E = in_sizes[1];
  const int U = in_sizes[23];
  const int C = in_sizes[20];

  // lower halves of the 2H x H gate weights (row-major: rows H..2H-1)
  const float* lzWb = lz_W + HD * HD;
  const float* lrWb = lr_W + HD * HD;
  const float* lhWb = lh_W + HD * HD;

  // workspace layout (floats)
  float* ws = (float*)d_ws;
  auto al = [](size_t x) { return (x + 63) & ~(size_t)63; };
  size_t o_deg    = 0;
  size_t o_pooled = al((size_t)N);
  size_t o_cz     = o_pooled + HD;
  size_t o_cr     = o_cz + HD;
  size_t o_ch     = o_cr + HD;
  size_t o_Mz     = al(o_ch + HD);
  size_t o_Mr     = o_Mz + HD * HD;
  size_t o_Mh     = o_Mr + HD * HD;
  size_t o_agg    = al(o_Mh + HD * HD);
  size_t o_z      = o_agg + (size_t)N * HD;
  size_t o_hr     = o_z + (size_t)N * HD;
  float* deg    = ws + o_deg;
  float* pooled = ws + o_pooled;
  float* cz = ws + o_cz; float* cr = ws + o_cr; float* ch = ws + o_ch;
  float* Mz = ws + o_Mz; float* Mr = ws + o_Mr; float* Mh = ws + o_Mh;
  float* agg = ws + o_agg;
  float* zbuf = ws + o_z;
  float* hrbuf = ws + o_hr;
  (void)ws_size; (void)n_in; (void)out_size;

  float* outp = (float*)d_out;
  float* pred = outp;          // [C]
  float* hnew = outp + C;      // [N, H]

  const long totalNH = (long)N * HD;

  // 1) deg = 1 (self loop), pooled = 0
  k_init<<<(N + 255) / 256, 256, 0, stream>>>(deg, pooled, N);
  // 2) deg += w over edges
  k_deg<<<(E + 255) / 256, 256, 0, stream>>>(dst, edge_w, deg, E);
  // 3) dinv
  k_dinv<<<(N + 255) / 256, 256, 0, stream>>>(deg, N);
  // 4) agg = dinv^2 * X  (self-loop contribution)
  k_selfloop<<<(int)((totalNH + 255) / 256), 256, 0, stream>>>(node_feat, deg, agg, totalNH);
  // 5) agg += A_norm-edges @ X   (32 lanes / edge)
  {
    long th = (long)E * 32;
    k_scatter<<<(int)((th + 255) / 256), 256, 0, stream>>>(node_feat, src, dst, edge_w, deg, agg, E);
  }
  // 6) fold weights
  k_prep<<<HD, HD, 0, stream>>>(Wz, bz, lz_W, lz_b, Mz, cz);
  k_prep<<<HD, HD, 0, stream>>>(Wr, br, lr_W, lr_b, Mr, cr);
  k_prep<<<HD, HD, 0, stream>>>(Wh, bh, lh_W, lh_b, Mh, ch);
  // 7) z, r gates (WMMA f32)
  k_gates_zr<<<N / 16, 256, 0, stream>>>(agg, h0, Mz, Mr, lzWb, lrWb, cz, cr, zbuf, hrbuf);
  // 8) candidate + combine -> h_new directly into d_out
  k_cand<<<N / 16, 256, 0, stream>>>(agg, hrbuf, Mh, lhWb, ch, zbuf, h0, hnew);
  // 9) pooled = sum_u relu(h_new[ids])
  k_pool<<<(U + 127) / 128, 128, 0, stream>>>(hnew, local_nids, pooled, U);
  // 10) head -> pred
  k_head<<<1, HD, 0, stream>>>(pooled, lin_W, lin_b, dn_W, dn_b, out_W, out_b,
                               pred, 1.0f / (float)U, C);
}